// BSLoss_13640816132730
// MI455X (gfx1250) — compile-verified
//
#include <hip/hip_runtime.h>
#include <math.h>

// Black-Scholes PINN loss, single streaming pass over V[4096][4096] (f32).
// HBM floor: 64MB @ 23.3 TB/s ~ 2.9us; 192MB L2 absorbs all stencil reuse,
// so minimize per-element VALU:
//  - residual refactored to row-uniform coefficients (1 reciprocal/row;
//    uniform math scalarizes to SALU, confirmed in round-1/2 asm)
//  - 8 cols/thread via two b128 loads + 2 edge scalars
//  - per-column accumulators + 0/1 weights: unconditional v_fmac in the hot
//    loop (no per-element cmp/cndmask), 8 independent FMA chains
//  - wave reduction via v_wmma_f32_16x16x4_f32 half-wave fold + 4 xor shuffles

#define NSS 4096        // rows (S axis)
#define NTT 4096        // cols (t axis, contiguous)
#define RPT 8           // rows per thread
#define CPT 8           // cols per thread

typedef __attribute__((ext_vector_type(2))) float v2f;
typedef __attribute__((ext_vector_type(8))) float v8f;

#define C_ALPHA   2.5f                 // 2r/sigma^2
#define C_INV2DU  2047.5f              // 1/(2*DU), DU = 1/4095
#define C_INVDU2  16769025.0f          // 1/DU^2 = 4095^2
#define C_INV2DT  102375.0f            // 1/(2*DT_NORM), DT_NORM = 0.02/4095
#define C_NINT    16760836.0f          // 4094*4094

__device__ __forceinline__ float waveSum32(float v) {
    v += __shfl_xor(v, 16);
    v += __shfl_xor(v, 8);
    v += __shfl_xor(v, 4);
    v += __shfl_xor(v, 2);
    v += __shfl_xor(v, 1);
    return v;
}

__device__ __forceinline__ void loadRow8(const float* __restrict__ p, float* r) {
    const float4* q = reinterpret_cast<const float4*>(p);   // 32B aligned (base%8==0)
    const float4 a = q[0];
    const float4 b = q[1];
    r[0] = a.x; r[1] = a.y; r[2] = a.z; r[3] = a.w;
    r[4] = b.x; r[5] = b.y; r[6] = b.z; r[7] = b.w;
}

__global__ __launch_bounds__(256)
void BSLoss_pde_kernel(const float* __restrict__ V, float* __restrict__ partials,
                       float C1, float C2)
{
    const int  ct   = blockIdx.x * blockDim.x + threadIdx.x;   // column-tile id
    const int  base = ct * CPT;                                // 0..4088, 32B aligned
    const int  i0   = 1 + blockIdx.y * RPT;                    // first interior row
    const float dL  = C2 - C1;
    const bool hasL = (base > 0);
    const bool hasR = (base + CPT < NTT);

    // 0/1 interior-column weights, applied once at the end (edge threads only
    // ever zero out col 0 / col 4095 contributions; masked residuals are finite)
    float w[CPT];
    #pragma unroll
    for (int k = 0; k < CPT; ++k) {
        const int c = base + k;
        w[k] = (c >= 1 && c <= NTT - 2) ? 1.0f : 0.0f;
    }

    float cm[CPT], cc[CPT], cp[CPT];
    const float* rowM = V + (size_t)(i0 - 1) * NTT + base;
    const float* rowC = V + (size_t)(i0    ) * NTT + base;
    loadRow8(rowM, cm);
    loadRow8(rowC, cc);
    float lf_c = hasL ? rowC[-1]  : 0.0f;   // t-neighbors of the current row
    float rt_c = hasR ? rowC[CPT] : 0.0f;

    float accv[CPT];
    #pragma unroll
    for (int k = 0; k < CPT; ++k) accv[k] = 0.0f;

    #pragma unroll
    for (int r = 0; r < RPT; ++r) {
        const int i = i0 + r;
        if (i > NSS - 2) break;
        const float* rowP = V + (size_t)(i + 1) * NTT + base;
        loadRow8(rowP, cp);
        const float lf_p = hasL ? rowP[-1]  : 0.0f;
        const float rt_p = hasR ? rowP[CPT] : 0.0f;

        // ---- row-uniform coefficients (SALU + one VALU reciprocal) ----
        const float u   = (float)i * (1.0f / 4095.0f);
        const float L   = fmaf(dL, u, C1);
        const float L2  = L * L;
        const float sn  = (100.0f + 30.0f * (L * L2 * (1.0f / 6.0f) + L)) * (1.0f / 300.0f);
        const float su  = (30.0f * dL * fmaf(0.5f, L2, 1.0f)) * (1.0f / 300.0f);
        const float suu = (30.0f * dL * dL * L) * (1.0f / 300.0f);
        const float isu = 1.0f / su;
        const float is2 = isu * isu;
        const float P   = C_INVDU2 * is2;                   // Vuu/su^2
        const float Q   = -(C_INV2DU * suu) * (is2 * isu);  // -Vu*suu/su^3
        const float M2P = -(P + P);
        const float NS2 = -(sn * sn);
        const float NRc = -(C_ALPHA * sn * C_INV2DU) * isu; // -alpha*sn/su

        #pragma unroll
        for (int k = 0; k < CPT; ++k) {
            const float lf = (k == 0)       ? lf_c : cc[k - 1];
            const float rt = (k == CPT - 1) ? rt_c : cc[k + 1];
            const float t1 = cp[k] - cm[k];
            const float t2 = cp[k] + cm[k];
            float VSS = fmaf(P, t2, fmaf(Q, t1, M2P * cc[k]));
            VSS = fminf(fmaxf(VSS, -100.0f), 100.0f);       // v_med3
            float res = fmaf(NS2, VSS, C_INV2DT * (rt - lf));
            res = fmaf(NRc, t1, res);
            res = fmaf(C_ALPHA, cc[k], res);
            accv[k] = fmaf(res, res, accv[k]);              // v_fmac, 8 chains
        }

        lf_c = lf_p; rt_c = rt_p;
        #pragma unroll
        for (int k = 0; k < CPT; ++k) { cm[k] = cc[k]; cc[k] = cp[k]; }
    }

    // fold per-column chains with interior weights (once per thread)
    float acc = 0.0f;
    #pragma unroll
    for (int k = 0; k < CPT; ++k) acc = fmaf(accv[k], w[k], acc);

    // ---- wave reduction: WMMA half-wave fold + 4-step xor tree ----
    // A = ones(16x4), B carries each lane's partial; D[m,n] = acc[n]+acc[n+16]
    // for either K->lane-half convention (we sum over all K). EXEC all ones.
    v2f a; a[0] = 1.0f; a[1] = 1.0f;
    v2f b; b[0] = acc;  b[1] = 0.0f;
    v8f c = {};
    v8f d = __builtin_amdgcn_wmma_f32_16x16x4_f32(
        false, a, false, b, (short)0, c, false, false);
    float s = d[0];
    s += __shfl_xor(s, 1);
    s += __shfl_xor(s, 2);
    s += __shfl_xor(s, 4);
    s += __shfl_xor(s, 8);

    __shared__ float lds[8];
    const int lane = threadIdx.x & 31;
    const int wv   = threadIdx.x >> 5;
    if (lane == 0) lds[wv] = s;
    __syncthreads();
    if (threadIdx.x == 0) {
        float t = 0.0f;
        #pragma unroll
        for (int w8 = 0; w8 < 8; ++w8) t += lds[w8];
        partials[blockIdx.y * gridDim.x + blockIdx.x] = t;
    }
}

__global__ __launch_bounds__(256)
void BSLoss_finish_kernel(const float* __restrict__ V,
                          const float* __restrict__ partials, int nPartials,
                          float* __restrict__ out)
{
    const int tid = threadIdx.x;
    float accP = 0.0f, accB = 0.0f, accT = 0.0f;

    for (int p = tid; p < nPartials; p += 256) accP += partials[p];

    // far-field BC at S = S_max (last row), over t
    for (int jj = tid; jj < NTT; jj += 256) {
        const float t      = (float)jj * (1.0f / 4095.0f);
        const float target = 1.0f - 100.0f * __expf(-0.05f * (1.0f - t)) * (1.0f / 300.0f);
        const float dfr    = V[(size_t)(NSS - 1) * NTT + jj] - target;
        accB += dfr * dfr;
    }

    // terminal Huber loss at t = T (last col), over u
    for (int ii = tid; ii < NSS; ii += 256) {
        const float u   = (float)ii * (1.0f / 4095.0f);
        const float x   = 50.0f * (u - (100.0f / 300.0f));
        const float sp  = (fmaxf(x, 0.0f) + log1pf(__expf(-fabsf(x)))) * (1.0f / 50.0f);
        const float dfr = V[(size_t)ii * NTT + (NTT - 1)] - sp;
        const float ad  = fabsf(dfr);
        accT += (ad < 0.01f) ? 0.5f * dfr * dfr : 0.01f * (ad - 0.005f);
    }

    accP = waveSum32(accP);
    accB = waveSum32(accB);
    accT = waveSum32(accT);

    __shared__ float sP[8], sB[8], sT[8];
    const int lane = tid & 31, wv = tid >> 5;
    if (lane == 0) { sP[wv] = accP; sB[wv] = accB; sT[wv] = accT; }
    __syncthreads();
    if (tid == 0) {
        float tp = 0.0f, tb = 0.0f, tt = 0.0f;
        #pragma unroll
        for (int w8 = 0; w8 < 8; ++w8) { tp += sP[w8]; tb += sB[w8]; tt += sT[w8]; }
        out[0] = tp * (1.0f / C_NINT)                 // LAMBDA_PDE = 1
               + 10.0f * tb * (1.0f / (float)NTT)     // LAMBDA_BC
               + 10.0f * tt * (1.0f / (float)NSS);    // LAMBDA_TC
    }
}

static double solve_depressed_cubic_host(double Q) {
    const double p = 6.0;                 // CHI
    const double q = 6.0 * Q;
    const double sp = sqrt(p);
    double arg = fabs(q) / (2.0 * p * sp / (3.0 * sqrt(3.0)));
    if (arg < 1.0) arg = 1.0;
    const double c = 2.0 * sp * cosh(acosh(arg) / 3.0);
    return (q >= 0.0) ? -c : c;
}

extern "C" void kernel_launch(void* const* d_in, const int* in_sizes, int n_in,
                              void* d_out, int out_size, void* d_ws, size_t ws_size,
                              hipStream_t stream) {
    const float* V  = (const float*)d_in[0];
    float* out      = (float*)d_out;
    float* partials = (float*)d_ws;

    const float C1 = (float)solve_depressed_cubic_host(100.0 / 30.0);   // (B-0)/30
    const float C2 = (float)solve_depressed_cubic_host(-200.0 / 30.0);  // (B-Smax)/30

    dim3 block(256);
    // cols: 4096/8 = 512 col-threads -> 2 blocks; rows: ceil(4094/8) = 512
    dim3 grid((NTT / CPT) / 256, (NSS - 2 + RPT - 1) / RPT);

    BSLoss_pde_kernel<<<grid, block, 0, stream>>>(V, partials, C1, C2);

    const int nP = (int)(grid.x * grid.y);
    BSLoss_finish_kernel<<<1, 256, 0, stream>>>(V, partials, nP, out);
}